// GraphSAGELayer_80934363726330
// MI455X (gfx1250) — compile-verified
//
#include <hip/hip_runtime.h>
#include <hip/hip_bf16.h>

typedef float v2f __attribute__((ext_vector_type(2)));
typedef float v8f __attribute__((ext_vector_type(8)));

#define D 128

// ---------------------------------------------------------------------------
// Kernel 1: zero the scatter workspace (nsum + counts)
// ---------------------------------------------------------------------------
__global__ void sage_zero_kernel(float* __restrict__ ws, long long total) {
    long long i = (long long)blockIdx.x * blockDim.x + threadIdx.x;
    long long stride = (long long)gridDim.x * blockDim.x;
    for (; i < total; i += stride) ws[i] = 0.0f;
}

// ---------------------------------------------------------------------------
// Kernel 2: edge scatter-add. One wave32 per edge; lane l handles features
// [4l, 4l+4). 4 global_atomic_add_f32 per lane, count bump on lane 0.
// ---------------------------------------------------------------------------
__global__ __launch_bounds__(256) void sage_scatter_kernel(
    const float* __restrict__ x,
    const long long* __restrict__ esrc,
    const long long* __restrict__ edst,
    float* __restrict__ nsum,
    float* __restrict__ counts,
    int n_edges)
{
    int wave = (int)((blockIdx.x * (unsigned)blockDim.x + threadIdx.x) >> 5);
    int lane = threadIdx.x & 31;
    if (wave >= n_edges) return;

    long long s = esrc[wave];
    long long d = edst[wave];

    const float4 v = *(const float4*)(x + s * D + lane * 4);
    float* p = nsum + d * D + lane * 4;
    atomicAdd(p + 0, v.x);
    atomicAdd(p + 1, v.y);
    atomicAdd(p + 2, v.z);
    atomicAdd(p + 3, v.w);
    if (lane == 0) atomicAdd(counts + d, 1.0f);
}

// ---------------------------------------------------------------------------
// Kernel 3: inv[i] = 1 / max(counts[i], 1)
// ---------------------------------------------------------------------------
__global__ void sage_inv_kernel(const float* __restrict__ counts,
                                float* __restrict__ inv, int n) {
    int i = blockIdx.x * blockDim.x + threadIdx.x;
    if (i < n) inv[i] = 1.0f / fmaxf(counts[i], 1.0f);
}

// ---------------------------------------------------------------------------
// Kernel 4: out = x @ Wself^T + b + (nsum*inv) @ Wneigh^T using
// V_WMMA_F32_16X16X4_F32. Block = 8 waves; block owns 16 output rows,
// wave w owns columns [16w, 16w+16).
//
// Fragment layouts (CDNA5 ISA 7.12.2, wave32):
//   A 16x4 f32 : lanes 0-15 -> M=lane,  {K=k0, k0+1}; lanes 16-31 -> {k0+2, k0+3}
//   B 4x16 f32 : lanes 0-15 -> N=lane,  {K=k0, k0+1}; lanes 16-31 -> {k0+2, k0+3}
//                with B[k][n] = W[n][k]  (out = in @ W^T)
//   C/D 16x16  : VGPR r, lanes 0-15 -> (M=r,   N=lane)
//                         lanes 16-31 -> (M=r+8, N=lane-16)
// ---------------------------------------------------------------------------
__global__ __launch_bounds__(256) void sage_gemm_kernel(
    const float* __restrict__ x,
    const float* __restrict__ nsum,
    const float* __restrict__ inv,
    const float* __restrict__ Wself,
    const float* __restrict__ Wneigh,
    const float* __restrict__ bias,
    float* __restrict__ out,
    int nodes)
{
    const int wave = threadIdx.x >> 5;
    const int lane = threadIdx.x & 31;
    const int half = lane >> 4;      // 0: lanes 0-15, 1: lanes 16-31
    const int l15  = lane & 15;

    const int m0 = blockIdx.x * 16;
    const int n0 = wave * 16;

    int row = m0 + l15;
    if (row > nodes - 1) row = nodes - 1;   // clamp: EXEC must stay all-ones
    const int ncol = n0 + l15;

    const float invr = inv[row];
    const float* __restrict__ xr = x     + (size_t)row  * D;
    const float* __restrict__ nr = nsum  + (size_t)row  * D;
    const float* __restrict__ ws = Wself + (size_t)ncol * D;
    const float* __restrict__ wn = Wneigh+ (size_t)ncol * D;

    v8f c = {};

#pragma unroll 8
    for (int k0 = 0; k0 < D; k0 += 4) {
        const int kb = k0 + 2 * half;

        const v2f a_s = *(const v2f*)(xr + kb);
        const v2f b_s = *(const v2f*)(ws + kb);
        c = __builtin_amdgcn_wmma_f32_16x16x4_f32(
                false, a_s, false, b_s, (short)0, c, false, false);

        v2f a_n = *(const v2f*)(nr + kb);
        a_n = a_n * invr;
        const v2f b_n = *(const v2f*)(wn + kb);
        c = __builtin_amdgcn_wmma_f32_16x16x4_f32(
                false, a_n, false, b_n, (short)0, c, false, false);
    }

    const float bv = bias[ncol];
#pragma unroll
    for (int r = 0; r < 8; ++r) {
        const int orow = m0 + r + 8 * half;
        if (orow < nodes)
            out[(size_t)orow * D + ncol] = c[r] + bv;
    }
}

// ---------------------------------------------------------------------------
// Host-side launcher
// ---------------------------------------------------------------------------
extern "C" void kernel_launch(void* const* d_in, const int* in_sizes, int n_in,
                              void* d_out, int out_size, void* d_ws, size_t ws_size,
                              hipStream_t stream) {
    const float*     x   = (const float*)d_in[0];
    const long long* ei  = (const long long*)d_in[1];   // [2, E] int64
    const float*     Wn  = (const float*)d_in[2];       // W_neigh
    const float*     Wsf = (const float*)d_in[3];       // W_self
    const float*     bs  = (const float*)d_in[4];       // b_self
    float*           out = (float*)d_out;

    const int nodes   = in_sizes[0] / D;
    const int n_edges = in_sizes[1] / 2;

    // workspace layout: nsum [nodes*D] | counts [nodes] | inv [nodes]
    float* nsum   = (float*)d_ws;
    float* counts = nsum + (size_t)nodes * D;
    float* inv    = counts + nodes;

    const long long zero_total = (long long)nodes * D + nodes;
    sage_zero_kernel<<<2048, 256, 0, stream>>>(nsum, zero_total);

    const int scat_blocks = (n_edges + 7) / 8;   // 8 waves (edges) per block
    sage_scatter_kernel<<<scat_blocks, 256, 0, stream>>>(
        x, ei, ei + n_edges, nsum, counts, n_edges);

    sage_inv_kernel<<<(nodes + 255) / 256, 256, 0, stream>>>(counts, inv, nodes);

    const int gemm_blocks = (nodes + 15) / 16;
    sage_gemm_kernel<<<gemm_blocks, 256, 0, stream>>>(
        x, nsum, inv, Wsf, Wn, bs, out, nodes);
}